// GraphSearch_5196910428568
// MI455X (gfx1250) — compile-verified
//
#include <hip/hip_runtime.h>
#include <hip/hip_bf16.h>

#define N_ENTITY 200000
#define EMBED 64
#define N_EDGES 6400000
#define BATCH 1024
#define Q_LEN 20
#define N_NEG 5
#define HEADS 4
#define HEAD_DIM 16
#define CONV_NUM 3

typedef float v2f __attribute__((ext_vector_type(2)));
typedef float v8f __attribute__((ext_vector_type(8)));

// ---------------- utility kernels ----------------

__global__ void fill_kernel(float* __restrict__ p, float v, long long n) {
    long long i = (long long)blockIdx.x * blockDim.x + threadIdx.x;
    long long stride = (long long)gridDim.x * blockDim.x;
    for (; i < n; i += stride) p[i] = v;
}

__global__ void copy_kernel(float* __restrict__ dst, const float* __restrict__ src,
                            long long n) {
    long long i = (long long)blockIdx.x * blockDim.x + threadIdx.x;
    long long stride = (long long)gridDim.x * blockDim.x;
    for (; i < n; i += stride) dst[i] = src[i];
}

// ---------------- SpMM: h_dst[row] += val * h_src[col] ----------------
// One wave32 per edge; each lane owns 2 consecutive floats of the 64-dim row.
__global__ void spmm_kernel(const float* __restrict__ hsrc, float* __restrict__ hdst,
                            const int* __restrict__ erow, const int* __restrict__ ecol,
                            const float* __restrict__ eval, int nEdges) {
    int wavesPerBlock = blockDim.x >> 5;
    long long e = (long long)blockIdx.x * wavesPerBlock + (threadIdx.x >> 5);
    if (e >= nEdges) return;
    int lane = threadIdx.x & 31;
    int c = ecol[e];
    int r = erow[e];
    float v = eval[e];
    const float2 x = *(const float2*)(hsrc + (size_t)c * EMBED + lane * 2);
    float* d = hdst + (size_t)r * EMBED + lane * 2;
    atomicAdd(d + 0, x.x * v);
    atomicAdd(d + 1, x.y * v);
}

// ---------------- gather rows needed by losses, accumulate over layers ----------------
// acc layout: [users(1024) | items(1024) | negs(5120)] x 64
__global__ void gather_acc_kernel(const float* __restrict__ src,
                                  const int* __restrict__ users,
                                  const int* __restrict__ items,
                                  const int* __restrict__ negs,
                                  float* __restrict__ acc, int init) {
    int i = blockIdx.x * blockDim.x + threadIdx.x;
    const int total = (2 * BATCH + BATCH * N_NEG) * EMBED;
    if (i >= total) return;
    int r = i >> 6;
    int d = i & 63;
    int idx;
    if (r < BATCH)          idx = users[r];
    else if (r < 2 * BATCH) idx = items[r - BATCH];
    else                    idx = negs[r - 2 * BATCH];
    float v = src[(size_t)idx * EMBED + d];
    if (init) acc[i] = v; else acc[i] += v;
}

// ---------------- word embedding gather ----------------
__global__ void gather_words_kernel(const float* __restrict__ word_w,
                                    const int* __restrict__ qw,
                                    float* __restrict__ x) {
    long long i = (long long)blockIdx.x * blockDim.x + threadIdx.x;
    if (i >= (long long)BATCH * Q_LEN * EMBED) return;
    long long m = i >> 6;
    int d = (int)(i & 63);
    x[i] = word_w[(size_t)qw[m] * EMBED + d];
}

// ---------------- WMMA f32 GEMM: C[M,N] = A[M,K] * W[N,K]^T + bias ----------------
// One wave per 16x16 output tile; K-loop in steps of 4 via V_WMMA_F32_16X16X4_F32.
__global__ void gemm_wmma_kernel(const float* __restrict__ A,
                                 const float* __restrict__ W,
                                 const float* __restrict__ bias,
                                 float* __restrict__ C,
                                 int M, int N, int K) {
    int wavesPerBlock = blockDim.x >> 5;
    int wave = blockIdx.x * wavesPerBlock + (threadIdx.x >> 5);
    int lane = threadIdx.x & 31;
    int tilesN = N >> 4;
    int tM = (wave / tilesN) << 4;
    int tN = (wave % tilesN) << 4;
    if (tM >= M) return;                 // wave-uniform: EXEC stays full for WMMA
    int half = lane >> 4;                // 0: lanes 0-15 (K+0,K+1), 1: lanes 16-31 (K+2,K+3)
    int r = lane & 15;
    const float* arow = A + (size_t)(tM + r) * K + (half << 1);
    const float* brow = W + (size_t)(tN + r) * K + (half << 1);  // B[k][n] = W[n][k]
    v8f c = {};
    for (int k = 0; k < K; k += 4) {
        v2f a = *(const v2f*)(arow + k);
        v2f b = *(const v2f*)(brow + k);
        c = __builtin_amdgcn_wmma_f32_16x16x4_f32(
                false, a, false, b, (short)0, c, false, false);
    }
    float bv = bias[tN + r];
    int mBase = tM + (half << 3);        // C/D: VGPR v holds M = v (+8 for lanes 16-31)
#pragma unroll
    for (int v = 0; v < 8; ++v)
        C[(size_t)(mBase + v) * N + (tN + r)] = c[v] + bv;
}

// ---------------- tiny multihead attention (heads=4, dim=16, seq=20) ----------------
__global__ void attn_kernel(const float* __restrict__ qkv, float* __restrict__ ctx) {
    __shared__ float sh[Q_LEN * 3 * EMBED];   // 20*192 floats = 15 KB
    int b = blockIdx.x;
    const float* src = qkv + (size_t)b * Q_LEN * 3 * EMBED;
    for (int i = threadIdx.x; i < Q_LEN * 3 * EMBED; i += blockDim.x) sh[i] = src[i];
    __syncthreads();
    int t = threadIdx.x;
    if (t < HEADS * Q_LEN) {
        int h = t / Q_LEN;
        int qi = t % Q_LEN;
        const float* qv = sh + qi * 3 * EMBED + h * HEAD_DIM;
        float s[Q_LEN];
        float mx = -1e30f;
#pragma unroll
        for (int j = 0; j < Q_LEN; ++j) {
            const float* kv = sh + j * 3 * EMBED + EMBED + h * HEAD_DIM;
            float acc = 0.f;
#pragma unroll
            for (int d = 0; d < HEAD_DIM; ++d) acc += qv[d] * kv[d];
            acc *= 0.25f;                 // 1/sqrt(HEAD_DIM)
            s[j] = acc;
            mx = fmaxf(mx, acc);
        }
        float sum = 0.f;
#pragma unroll
        for (int j = 0; j < Q_LEN; ++j) { s[j] = expf(s[j] - mx); sum += s[j]; }
        float inv = 1.0f / sum;
        float* out = ctx + ((size_t)b * Q_LEN + qi) * EMBED + h * HEAD_DIM;
#pragma unroll
        for (int d = 0; d < HEAD_DIM; ++d) {
            float acc = 0.f;
#pragma unroll
            for (int j = 0; j < Q_LEN; ++j)
                acc += s[j] * sh[j * 3 * EMBED + 2 * EMBED + h * HEAD_DIM + d];
            out[d] = acc * inv;
        }
    }
}

// ---------------- mean over Q ----------------
__global__ void mean_kernel(const float* __restrict__ sout, float* __restrict__ qenc) {
    int i = blockIdx.x * blockDim.x + threadIdx.x;
    if (i >= BATCH * EMBED) return;
    int b = i >> 6;
    int d = i & 63;
    float acc = 0.f;
    for (int q = 0; q < Q_LEN; ++q) acc += sout[((size_t)b * Q_LEN + q) * EMBED + d];
    qenc[i] = acc * (1.0f / Q_LEN);
}

// ---------------- losses ----------------
__device__ __forceinline__ float softplus(float z) {  // log(1+exp(z)), stable
    if (z > 20.f) return z;
    if (z < -20.f) return expf(z);
    return log1pf(expf(z));
}

__global__ void loss_kernel(const float* __restrict__ acc,
                            const float* __restrict__ qenc,
                            float* __restrict__ partial) {
    int b = blockIdx.x * blockDim.x + threadIdx.x;
    if (b >= BATCH) return;
    const float* aU = acc + (size_t)b * EMBED;
    const float* aI = acc + (size_t)(BATCH + b) * EMBED;
    const float* aN = acc + (size_t)(2 * BATCH) * EMBED + (size_t)b * N_NEG * EMBED;
    const float* qe = qenc + (size_t)b * EMBED;
    const float scale = 1.0f / (CONV_NUM + 1);
    float dui = 0.f, dpi = 0.f;
    float dun[N_NEG] = {0, 0, 0, 0, 0};
    float dpn[N_NEG] = {0, 0, 0, 0, 0};
    for (int d = 0; d < EMBED; ++d) {
        float u = scale * aU[d];
        float it = scale * aI[d];
        float p = qe[d] + 0.1f * u;
        dui += u * it;
        dpi += p * it;
#pragma unroll
        for (int k = 0; k < N_NEG; ++k) {
            float nv = scale * aN[k * EMBED + d];
            dun[k] += u * nv;
            dpn[k] += p * nv;
        }
    }
    float cf = 0.f, sneg = 0.f;
#pragma unroll
    for (int k = 0; k < N_NEG; ++k) {
        cf += softplus(dun[k] - dui);   // -logsig(s_pos - s_neg)
        sneg += softplus(dpn[k]);       // -logsig(-s_neg)
    }
    partial[b] = cf * (1.0f / (BATCH * N_NEG))
               + softplus(-dpi) * (1.0f / BATCH)
               + sneg * (1.0f / (BATCH * N_NEG));
}

__global__ void reduce_kernel(const float* __restrict__ partial, float* __restrict__ out) {
    __shared__ float sh[256];
    float acc = 0.f;
    for (int i = threadIdx.x; i < BATCH; i += 256) acc += partial[i];
    sh[threadIdx.x] = acc;
    __syncthreads();
    for (int s = 128; s > 0; s >>= 1) {
        if (threadIdx.x < s) sh[threadIdx.x] += sh[threadIdx.x + s];
        __syncthreads();
    }
    if (threadIdx.x == 0) out[0] = sh[0];
}

// ---------------- host orchestration ----------------

extern "C" void kernel_launch(void* const* d_in, const int* in_sizes, int n_in,
                              void* d_out, int out_size, void* d_ws, size_t ws_size,
                              hipStream_t stream) {
    const float* entity_w   = (const float*)d_in[0];
    const float* word_w     = (const float*)d_in[1];
    const float* in_proj_w  = (const float*)d_in[2];
    const float* in_proj_b  = (const float*)d_in[3];
    const float* out_proj_w = (const float*)d_in[4];
    const float* out_proj_b = (const float*)d_in[5];
    const float* edge_val   = (const float*)d_in[6];
    const int*   users      = (const int*)d_in[7];
    const int*   items      = (const int*)d_in[8];
    const int*   qwords     = (const int*)d_in[9];
    const int*   negs       = (const int*)d_in[10];
    const int*   erow       = (const int*)d_in[11];
    const int*   ecol       = (const int*)d_in[12];
    float* out = (float*)d_out;

    float* ws = (float*)d_ws;
    size_t off = 0;
    float* h0   = ws + off; off += (size_t)N_ENTITY * EMBED;               // 51.2 MB
    float* h1   = ws + off; off += (size_t)N_ENTITY * EMBED;               // 51.2 MB
    float* acc  = ws + off; off += (size_t)(2 * BATCH + BATCH * N_NEG) * EMBED;
    float* xbuf = ws + off; off += (size_t)BATCH * Q_LEN * EMBED;
    float* qkv  = ws + off; off += (size_t)BATCH * Q_LEN * 3 * EMBED;
    float* ctx  = ws + off; off += (size_t)BATCH * Q_LEN * EMBED;
    float* qenc = ws + off; off += (size_t)BATCH * EMBED;
    float* part = ws + off; off += BATCH;
    float* sout = xbuf;  // xbuf dead after in_proj GEMM; reuse for out_proj result

    const long long nH = (long long)N_ENTITY * EMBED;
    const int gatherN = (2 * BATCH + BATCH * N_NEG) * EMBED;

    // layer 0 of graph embedding
    copy_kernel<<<2048, 256, 0, stream>>>(h0, entity_w, nH);
    gather_acc_kernel<<<(gatherN + 255) / 256, 256, 0, stream>>>(
        entity_w, users, items, negs, acc, 1);

    // 3 SpMM propagation layers, accumulating gathered rows after each
    float* cur = h0;
    float* nxt = h1;
    for (int layer = 0; layer < CONV_NUM; ++layer) {
        fill_kernel<<<2048, 256, 0, stream>>>(nxt, 0.0f, nH);
        spmm_kernel<<<N_EDGES / 8, 256, 0, stream>>>(cur, nxt, erow, ecol,
                                                     edge_val, N_EDGES);
        gather_acc_kernel<<<(gatherN + 255) / 256, 256, 0, stream>>>(
            nxt, users, items, negs, acc, 0);
        float* t = cur; cur = nxt; nxt = t;
    }

    // sentence encoder
    gather_words_kernel<<<((long long)BATCH * Q_LEN * EMBED + 255) / 256, 256, 0, stream>>>(
        word_w, qwords, xbuf);
    {
        int M = BATCH * Q_LEN, N = 3 * EMBED, K = EMBED;   // 20480 x 192 x 64
        int tiles = (M / 16) * (N / 16);                   // 15360 -> 1920 blocks
        gemm_wmma_kernel<<<tiles / 8, 256, 0, stream>>>(xbuf, in_proj_w, in_proj_b,
                                                        qkv, M, N, K);
    }
    attn_kernel<<<BATCH, 128, 0, stream>>>(qkv, ctx);
    {
        int M = BATCH * Q_LEN, N = EMBED, K = EMBED;       // 20480 x 64 x 64
        int tiles = (M / 16) * (N / 16);                   // 5120 -> 640 blocks
        gemm_wmma_kernel<<<tiles / 8, 256, 0, stream>>>(ctx, out_proj_w, out_proj_b,
                                                        sout, M, N, K);
    }
    mean_kernel<<<(BATCH * EMBED + 255) / 256, 256, 0, stream>>>(sout, qenc);

    // losses (deterministic two-stage reduction)
    loss_kernel<<<(BATCH + 255) / 256, 256, 0, stream>>>(acc, qenc, part);
    reduce_kernel<<<1, 256, 0, stream>>>(part, out);
}